// GraphChenn_20521353740692
// MI455X (gfx1250) — compile-verified
//
#include <hip/hip_runtime.h>
#include <hip/hip_bf16.h>
#include <math.h>

// ---------------------------------------------------------------------------
// GCNII layer: out = theta*(support@W) + (1-theta)*support + input
//   support = (1-alpha)*SpMM(adj, input) + alpha*h0
// N=50000 nodes, E=800000 edges, D=128 feats. fp32 throughout.
// ---------------------------------------------------------------------------

#define N_NODES 50000
#define N_EDGES 800000
#define D_FEAT  128
#define LDS_STRIDE 132   // 128 + 4 pad -> bank stride 4, conflict-free column reads

typedef __attribute__((ext_vector_type(2))) float v2f;
typedef __attribute__((ext_vector_type(8))) float v8f;

// ---------------- Kernel 1: zero the hi accumulator (in workspace) ----------
__global__ void zero_hi_kernel(float4* __restrict__ hi4, int n4) {
    int i = blockIdx.x * blockDim.x + threadIdx.x;
    int stride = gridDim.x * blockDim.x;
    float4 z = make_float4(0.f, 0.f, 0.f, 0.f);
    for (; i < n4; i += stride) hi4[i] = z;
}

// ---------------- Kernel 2: SpMM scatter  hi[row] += w * input[col] ---------
// One wave (32 lanes) per edge; each lane handles 4 contiguous features.
__global__ void spmm_edges_kernel(const float* __restrict__ inp,
                                  const int* __restrict__ erow,
                                  const int* __restrict__ ecol,
                                  const float* __restrict__ ew,
                                  float* __restrict__ hi) {
    int lane = threadIdx.x & 31;
    int e = blockIdx.x * (blockDim.x >> 5) + (threadIdx.x >> 5);
    if (e >= N_EDGES) return;
    int r = erow[e];
    int c = ecol[e];
    float w = ew[e];
    // 16B vector gather of the source row (L2-resident: input is 25.6MB << 192MB L2)
    const float4 v = *((const float4*)(inp + (size_t)c * D_FEAT) + lane);
    float* dst = hi + (size_t)r * D_FEAT + lane * 4;
    atomicAdd(dst + 0, w * v.x);
    atomicAdd(dst + 1, w * v.y);
    atomicAdd(dst + 2, w * v.z);
    atomicAdd(dst + 3, w * v.w);
}

// ---------------- Kernel 3: fused support + WMMA fp32 GEMM + epilogue -------
// Grid: 3125 blocks (16-row M tiles). Block: 256 threads = 8 waves.
// Wave w computes the 16x16 output tile at columns [16w, 16w+16).
__global__ void fused_gemm_kernel(const float* __restrict__ inp,
                                  const float* __restrict__ h0,
                                  const float* __restrict__ W,     // [128][128] row-major (k,n)
                                  const float* __restrict__ hi,
                                  const float* __restrict__ lamda_p,
                                  const float* __restrict__ alpha_p,
                                  const int*   __restrict__ l_p,
                                  float* __restrict__ out) {
    __shared__ float sm[16 * LDS_STRIDE];   // support tile, 16 rows x 128 cols (padded)

    const float alpha = alpha_p[0];
    const float lamda = lamda_p[0];
    const float theta = fminf(1.0f, logf(lamda / (float)l_p[0] + 1.0f));

    const int tid  = threadIdx.x;
    const int lane = tid & 31;
    const int wave = tid >> 5;              // 0..7 -> N tile
    const int row0 = blockIdx.x * 16;

    // Cooperative: support = (1-alpha)*hi + alpha*h0 into LDS (2048 elems, 8/thread)
    #pragma unroll
    for (int i = tid; i < 16 * D_FEAT; i += 256) {
        int r = i >> 7;          // /128
        int c = i & 127;
        size_t g = (size_t)(row0 + r) * D_FEAT + c;
        sm[r * LDS_STRIDE + c] = (1.0f - alpha) * hi[g] + alpha * h0[g];
    }
    __syncthreads();

    // WMMA f32 16x16x4 fragment coordinates (wave32):
    //   A (16x4): lanes 0-15 -> K={k,k+1}, lanes 16-31 -> K={k+2,k+3}; M = lane&15
    //   B (4x16): same K split; N = lane&15
    //   C/D: vgpr i -> row = 8*(lane>=16)+i, col = lane&15
    const int laneHi = lane >> 4;
    const int m      = lane & 15;
    const int ncol   = wave * 16 + (lane & 15);

    __builtin_prefetch(&W[ncol], 0, 1);     // warm weight columns (global_prefetch_b8)

    v8f acc = {};
    #pragma unroll
    for (int k = 0; k < D_FEAT; k += 4) {
        const int kb = k + 2 * laneHi;
        v2f a, b;
        a.x = sm[m * LDS_STRIDE + kb];
        a.y = sm[m * LDS_STRIDE + kb + 1];
        b.x = W[(size_t)kb * D_FEAT + ncol];
        b.y = W[(size_t)(kb + 1) * D_FEAT + ncol];
        // D = A x B + C   -> v_wmma_f32_16x16x4_f32
        acc = __builtin_amdgcn_wmma_f32_16x16x4_f32(
            /*neg_a=*/false, a, /*neg_b=*/false, b,
            /*c_mod=*/(short)0, acc, /*reuse_a=*/false, /*reuse_b=*/false);
    }

    // Epilogue: out = theta*acc + (1-theta)*support + input
    #pragma unroll
    for (int i = 0; i < 8; ++i) {
        int r  = (laneHi ? 8 : 0) + i;
        size_t g = (size_t)(row0 + r) * D_FEAT + ncol;
        float s = sm[r * LDS_STRIDE + ncol];
        out[g] = theta * acc[i] + (1.0f - theta) * s + inp[g];
    }
}

// ---------------------------------------------------------------------------
extern "C" void kernel_launch(void* const* d_in, const int* in_sizes, int n_in,
                              void* d_out, int out_size, void* d_ws, size_t ws_size,
                              hipStream_t stream) {
    const float* inp   = (const float*)d_in[0];
    const float* h0    = (const float*)d_in[1];
    const float* W     = (const float*)d_in[2];
    const int*   erow  = (const int*)d_in[3];
    const int*   ecol  = (const int*)d_in[4];
    const float* ew    = (const float*)d_in[5];
    const float* lamda = (const float*)d_in[6];
    const float* alpha = (const float*)d_in[7];
    const int*   lval  = (const int*)d_in[8];
    float* out = (float*)d_out;

    float* hi = (float*)d_ws;   // N_NODES * D_FEAT floats = 25.6 MB scratch

    // 1) zero hi
    {
        int n4 = (N_NODES * D_FEAT) / 4;
        int blocks = 2048;
        zero_hi_kernel<<<blocks, 256, 0, stream>>>((float4*)hi, n4);
    }
    // 2) edge scatter SpMM (8 edges per 256-thread block)
    {
        int blocks = (N_EDGES + 7) / 8;
        spmm_edges_kernel<<<blocks, 256, 0, stream>>>(inp, erow, ecol, ew, hi);
    }
    // 3) fused support + GEMM (WMMA fp32) + epilogue
    {
        int blocks = N_NODES / 16;   // 3125, exact
        fused_gemm_kernel<<<blocks, 256, 0, stream>>>(inp, h0, W, hi,
                                                      lamda, alpha, lval, out);
    }
}